// Phase10TinyAttentionModel_10136122819119
// MI455X (gfx1250) — compile-verified
//
#include <hip/hip_runtime.h>
#include <hip/hip_bf16.h>
#include <math.h>

typedef __attribute__((ext_vector_type(2))) float v2f;
typedef __attribute__((ext_vector_type(8))) float v8f;

#define D_DIM   1024
#define T_DIM   4096
#define B_DIM   8
#define NCLS    1000

// ---------------------------------------------------------------------------
// Small-M GEMM via V_WMMA_F32_16X16X4_F32.
// C[8 x N] = A[8 x K] * B (+bias). One wave (32 threads) per 16-wide N tile.
// M padded to 16; rows 8..15 zero-masked branchlessly so EXEC stays all-1s
// through the WMMA loop (ISA requirement).
// A-matrix 16x4 layout (ISA 7.12.2): lane%16 = M row; lane/16 selects K pair
//   (a.x = A[m][k+2*(lane/16)], a.y = A[m][k+2*(lane/16)+1]).
// B-matrix 4x16: row-striped across lanes: b.x = B[k+2*(lane/16)][n],
//   b.y = B[k+2*(lane/16)+1][n], n = lane%16.
// C/D 16x16: VGPR i, lanes 0-15 hold M=i (we only store M=0..7).
// ---------------------------------------------------------------------------
template <int TRANSB>
__global__ void gemm8_wmma(const float* __restrict__ A, long lda,
                           const float* __restrict__ Bm, long ldb,
                           const float* __restrict__ bias,
                           float* __restrict__ C, long ldc,
                           int N, int K) {
    const int lane = threadIdx.x;          // 0..31, one wave per block
    const int n0   = blockIdx.x * 16;
    const int m    = lane & 15;
    const int kk   = (lane >> 4) << 1;     // 0 or 2
    const int n    = n0 + (lane & 15);

    const int   nc    = (n < N) ? n : (N - 1);   // clamp -> no OOB loads
    const float nmask = (n < N) ? 1.0f : 0.0f;
    const int   mc    = m & 7;                   // clamp rows 8..15 -> 0..7
    const float amask = (m < 8) ? 1.0f : 0.0f;

    v8f acc = {};
    for (int k = 0; k < K; k += 4) {
        const int k0 = k + kk;
        v2f a;
        a.x = A[(size_t)mc * lda + k0]     * amask;
        a.y = A[(size_t)mc * lda + k0 + 1] * amask;
        v2f b;
        if (TRANSB) {   // B[k][n] = Bm[n*ldb + k]
            b.x = Bm[(size_t)nc * ldb + k0];
            b.y = Bm[(size_t)nc * ldb + k0 + 1];
        } else {        // B[k][n] = Bm[k*ldb + n]
            b.x = Bm[(size_t)k0 * ldb + nc];
            b.y = Bm[(size_t)(k0 + 1) * ldb + nc];
        }
        b.x *= nmask;
        b.y *= nmask;
        // (neg_a, A, neg_b, B, c_mod, C, reuse_a, reuse_b)
        acc = __builtin_amdgcn_wmma_f32_16x16x4_f32(
            false, a, false, b, (short)0, acc, false, false);
    }

    if (lane < 16 && n < N) {   // lanes 0-15 hold rows M=0..7 in acc[0..7]
        const float bb = bias ? bias[n] : 0.0f;
#pragma unroll
        for (int i = 0; i < 8; ++i)
            C[(size_t)i * ldc + n] = acc[i] + bb;
    }
}

// ---------------------------------------------------------------------------
// scores[b,t] = (1/32) * dot(qk[b,:], x[b,t,:])   — streams x (128 MB) once.
// One wave per (b,t) row: 8 float4 loads/lane, wave32 shuffle reduction.
// ---------------------------------------------------------------------------
__global__ void scores_kernel(const float* __restrict__ x,
                              const float* __restrict__ qk,
                              float* __restrict__ scores) {
    const int wave = (int)((blockIdx.x * blockDim.x + threadIdx.x) >> 5);
    const int lane = threadIdx.x & 31;
    const int b = wave >> 12;          // / 4096
    const int t = wave & (T_DIM - 1);

    const float4* xr = (const float4*)(x + ((size_t)b * T_DIM + t) * D_DIM);
    const float4* qr = (const float4*)(qk + (size_t)b * D_DIM);

    float4 acc = {0.f, 0.f, 0.f, 0.f};
#pragma unroll
    for (int j = 0; j < 8; ++j) {
        const float4 xv = xr[j * 32 + lane];
        const float4 qv = qr[j * 32 + lane];
        acc.x = fmaf(xv.x, qv.x, acc.x);
        acc.y = fmaf(xv.y, qv.y, acc.y);
        acc.z = fmaf(xv.z, qv.z, acc.z);
        acc.w = fmaf(xv.w, qv.w, acc.w);
    }
    float s = (acc.x + acc.y) + (acc.z + acc.w);
#pragma unroll
    for (int off = 16; off >= 1; off >>= 1)
        s += __shfl_xor(s, off, 32);
    if (lane == 0)
        scores[(size_t)b * T_DIM + t] = s * 0.03125f;  // 1/sqrt(1024)
}

// ---------------------------------------------------------------------------
// Softmax over T=4096 per batch. One block per b, 256 threads x 16 elems.
// ---------------------------------------------------------------------------
__global__ void softmax_kernel(const float* __restrict__ scores,
                               float* __restrict__ attn) {
    __shared__ float red[256];
    const int b = blockIdx.x;
    const int tid = threadIdx.x;
    const float* sc = scores + (size_t)b * T_DIM;
    float* at = attn + (size_t)b * T_DIM;

    float mx = -INFINITY;
#pragma unroll
    for (int i = 0; i < 16; ++i)
        mx = fmaxf(mx, sc[i * 256 + tid]);
    red[tid] = mx;
    __syncthreads();
    for (int s = 128; s > 0; s >>= 1) {
        if (tid < s) red[tid] = fmaxf(red[tid], red[tid + s]);
        __syncthreads();
    }
    const float M = red[0];
    __syncthreads();

    float sum = 0.f;
#pragma unroll
    for (int i = 0; i < 16; ++i) {
        const float e = __expf(sc[i * 256 + tid] - M);
        at[i * 256 + tid] = e;
        sum += e;
    }
    red[tid] = sum;
    __syncthreads();
    for (int s = 128; s > 0; s >>= 1) {
        if (tid < s) red[tid] += red[tid + s];
        __syncthreads();
    }
    const float inv = 1.0f / red[0];
#pragma unroll
    for (int i = 0; i < 16; ++i)
        at[i * 256 + tid] *= inv;
}

// ---------------------------------------------------------------------------
// partial[b,c,:] = sum over 64-row t-chunk c of attn[b,t] * x[b,t,:]
// (second streaming pass over x; x fits in 192 MB L2 so this should hit L2).
// 512 blocks x 256 threads; thread owns one float4 d-column. Deterministic.
// ---------------------------------------------------------------------------
__global__ void wsum_kernel(const float* __restrict__ x,
                            const float* __restrict__ attn,
                            float* __restrict__ partial) {
    const int b = blockIdx.x >> 6;      // / 64 chunks
    const int c = blockIdx.x & 63;
    const int tid = threadIdx.x;        // 0..255 -> float4 column

    const float4* xp =
        (const float4*)(x + ((size_t)b * T_DIM + (size_t)c * 64) * D_DIM);
    const float* ap = attn + (size_t)b * T_DIM + (size_t)c * 64;

    float4 acc = {0.f, 0.f, 0.f, 0.f};
    for (int t = 0; t < 64; ++t) {
        const float w = ap[t];
        const float4 xv = xp[t * 256 + tid];
        acc.x = fmaf(w, xv.x, acc.x);
        acc.y = fmaf(w, xv.y, acc.y);
        acc.z = fmaf(w, xv.z, acc.z);
        acc.w = fmaf(w, xv.w, acc.w);
    }
    ((float4*)(partial + ((size_t)(b * 64 + c) * D_DIM)))[tid] = acc;
}

// a[b,d] = sum_c partial[b,c,d]  (fixed order -> deterministic)
__global__ void reduce_kernel(const float* __restrict__ partial,
                              float* __restrict__ avec) {
    const int idx = blockIdx.x * 256 + threadIdx.x;  // 0..8191
    const int b = idx >> 10;
    const int d = idx & (D_DIM - 1);
    float s = 0.f;
#pragma unroll 8
    for (int c = 0; c < 64; ++c)
        s += partial[((size_t)(b * 64 + c) << 10) + d];
    avec[idx] = s;
}

// ---------------------------------------------------------------------------
extern "C" void kernel_launch(void* const* d_in, const int* in_sizes, int n_in,
                              void* d_out, int out_size, void* d_ws, size_t ws_size,
                              hipStream_t stream) {
    const float* x  = (const float*)d_in[0];
    const float* Wq = (const float*)d_in[1];
    const float* Wk = (const float*)d_in[2];
    const float* Wv = (const float*)d_in[3];
    const float* Wc = (const float*)d_in[4];
    const float* bc = (const float*)d_in[5];
    float* out = (float*)d_out;

    float* ws      = (float*)d_ws;
    float* q       = ws;            // 8*1024
    float* qk      = ws + 8192;     // 8*1024
    float* scores  = ws + 16384;    // 8*4096
    float* attn    = ws + 49152;    // 8*4096
    float* avec    = ws + 81920;    // 8*1024
    float* h       = ws + 90112;    // 8*1024
    float* partial = ws + 98304;    // 8*64*1024 = 2 MB

    const long ldd = D_DIM;

    // q = x[:,0,:] @ Wq           (A row stride = T*D)
    gemm8_wmma<0><<<64, 32, 0, stream>>>(x, (long)T_DIM * D_DIM, Wq, ldd,
                                         nullptr, q, ldd, D_DIM, D_DIM);
    // qk = q @ Wk^T               (folds Wk into the query)
    gemm8_wmma<1><<<64, 32, 0, stream>>>(q, ldd, Wk, ldd,
                                         nullptr, qk, ldd, D_DIM, D_DIM);
    // scores = (qk . x) / 32      (streaming pass 1 over x)
    scores_kernel<<<(B_DIM * T_DIM) / 8, 256, 0, stream>>>(x, qk, scores);
    // attn = softmax(scores)
    softmax_kernel<<<B_DIM, 256, 0, stream>>>(scores, attn);
    // a = attn-weighted sum of x rows (streaming pass 2, L2-resident)
    wsum_kernel<<<B_DIM * 64, 256, 0, stream>>>(x, attn, partial);
    reduce_kernel<<<32, 256, 0, stream>>>(partial, avec);
    // h = a @ Wv
    gemm8_wmma<0><<<64, 32, 0, stream>>>(avec, ldd, Wv, ldd,
                                         nullptr, h, ldd, D_DIM, D_DIM);
    // out = h @ Wc + bc           (N=1000, tile-edge masked)
    gemm8_wmma<0><<<63, 32, 0, stream>>>(h, ldd, Wc, (long)NCLS,
                                         bc, out, (long)NCLS, NCLS, D_DIM);
}